// AttentionBasedGNN_24249385353868
// MI455X (gfx1250) — compile-verified
//
#include <hip/hip_runtime.h>
#include <math.h>

// ---- problem constants (match reference) ----
#define N_NODES  50000
#define F_IN     64
#define NHEAD1   8
#define CH1      8
#define D1       64            // NHEAD1*CH1
#define NCLS     16
#define E_EDGES  800000
#define ET       (E_EDGES + N_NODES)   // + self loops
#define NEG_SLOPE 0.2f
#define ORD_NEG_INF 0x007FFFFFu        // order-encoding of -inf

typedef __attribute__((ext_vector_type(2))) float v2f;
typedef __attribute__((ext_vector_type(8))) float v8f;

// order-preserving float<->uint encoding for atomicMax on floats
__device__ __forceinline__ unsigned f2ord(float f) {
    unsigned u = __float_as_uint(f);
    return (u >> 31) ? ~u : (u | 0x80000000u);
}
__device__ __forceinline__ float ord2f(unsigned o) {
    return (o >> 31) ? __uint_as_float(o & 0x7FFFFFFFu) : __uint_as_float(~o);
}
__device__ __forceinline__ float lrelu(float v) { return v > 0.f ? v : NEG_SLOPE * v; }

// ---------------- init kernels ----------------
__global__ void init1_kernel(float* agg1, unsigned* m1, float* s1) {
    int i = blockIdx.x * blockDim.x + threadIdx.x;
    if (i < N_NODES * D1) agg1[i] = 0.f;
    if (i < N_NODES * NHEAD1) { m1[i] = ORD_NEG_INF; s1[i] = 0.f; }
}
__global__ void init2_kernel(float* dout, unsigned* m2, float* s2) {
    int i = blockIdx.x * blockDim.x + threadIdx.x;
    if (i < N_NODES * NCLS) dout[i] = 0.f;
    if (i < N_NODES) { m2[i] = ORD_NEG_INF; s2[i] = 0.f; }
}

// ---------------- GEMM1: h1[N,64] = x[N,64] @ W1[64,64]  (WMMA f32 16x16x4) ----
// One wave computes a 16x64 output tile; K=64 -> 16 wmma steps x 4 column tiles.
__global__ void __launch_bounds__(256) gemm1_wmma(const float* __restrict__ x,
                                                  const float* __restrict__ W,
                                                  float* __restrict__ h1) {
    __shared__ float sW[64 * 64];        // 16 KB, shared by block
    __shared__ float sX[8][16 * 64];     // 4 KB per wave
    const int tid = threadIdx.x;
    for (int i = tid; i < 64 * 64; i += 256) sW[i] = W[i];

    const int wave = tid >> 5, lane = tid & 31;
    const int row0 = (blockIdx.x * 8 + wave) * 16;
    float* sx = sX[wave];
    for (int i = lane; i < 16 * 64; i += 32) {
        int gr = row0 + (i >> 6);
        sx[i] = (gr < N_NODES) ? x[(size_t)gr * 64 + (i & 63)] : 0.f;
    }
    __syncthreads();
    if (row0 >= N_NODES) return;        // wave-uniform: EXEC stays all-1s for WMMA

    const int m  = lane & 15;           // A row / B,C column
    const int kh = (lane >> 4) << 1;    // K sub-offset: lanes 0-15 -> 0, 16-31 -> 2
    v8f acc0 = {}, acc1 = {}, acc2 = {}, acc3 = {};
#pragma unroll
    for (int kb = 0; kb < 16; ++kb) {
        const int k0 = kb * 4 + kh;
        v2f a;                          // 32-bit A 16x4 layout (ISA 7.12.2)
        a.x = sx[m * 64 + k0];
        a.y = sx[m * 64 + k0 + 1];
        v2f b0, b1, b2, b3;             // 32-bit B 4x16: row K striped over lanes
        b0.x = sW[k0 * 64 +  0 + m];  b0.y = sW[(k0 + 1) * 64 +  0 + m];
        b1.x = sW[k0 * 64 + 16 + m];  b1.y = sW[(k0 + 1) * 64 + 16 + m];
        b2.x = sW[k0 * 64 + 32 + m];  b2.y = sW[(k0 + 1) * 64 + 32 + m];
        b3.x = sW[k0 * 64 + 48 + m];  b3.y = sW[(k0 + 1) * 64 + 48 + m];
        acc0 = __builtin_amdgcn_wmma_f32_16x16x4_f32(false, a, false, b0, (short)0, acc0, false, false);
        acc1 = __builtin_amdgcn_wmma_f32_16x16x4_f32(false, a, false, b1, (short)0, acc1, false, false);
        acc2 = __builtin_amdgcn_wmma_f32_16x16x4_f32(false, a, false, b2, (short)0, acc2, false, false);
        acc3 = __builtin_amdgcn_wmma_f32_16x16x4_f32(false, a, false, b3, (short)0, acc3, false, false);
    }
    const int rbase = row0 + ((lane >> 4) << 3);   // hi lanes hold rows M+8
#pragma unroll
    for (int r = 0; r < 8; ++r) {
        size_t g = (size_t)(rbase + r) * 64 + m;
        h1[g +  0] = acc0[r];
        h1[g + 16] = acc1[r];
        h1[g + 32] = acc2[r];
        h1[g + 48] = acc3[r];
    }
}

// ---------------- GEMM2: h2[N,16] = out1[N,64] @ W2[64,16] ----------------
__global__ void __launch_bounds__(256) gemm2_wmma(const float* __restrict__ x,
                                                  const float* __restrict__ W,
                                                  float* __restrict__ h2) {
    __shared__ float sW[64 * 16];        // 4 KB
    __shared__ float sX[8][16 * 64];
    const int tid = threadIdx.x;
    for (int i = tid; i < 64 * 16; i += 256) sW[i] = W[i];

    const int wave = tid >> 5, lane = tid & 31;
    const int row0 = (blockIdx.x * 8 + wave) * 16;
    float* sx = sX[wave];
    for (int i = lane; i < 16 * 64; i += 32) {
        int gr = row0 + (i >> 6);
        sx[i] = (gr < N_NODES) ? x[(size_t)gr * 64 + (i & 63)] : 0.f;
    }
    __syncthreads();
    if (row0 >= N_NODES) return;

    const int m  = lane & 15;
    const int kh = (lane >> 4) << 1;
    v8f acc = {};
#pragma unroll
    for (int kb = 0; kb < 16; ++kb) {
        const int k0 = kb * 4 + kh;
        v2f a; a.x = sx[m * 64 + k0]; a.y = sx[m * 64 + k0 + 1];
        v2f b; b.x = sW[k0 * 16 + m]; b.y = sW[(k0 + 1) * 16 + m];
        acc = __builtin_amdgcn_wmma_f32_16x16x4_f32(false, a, false, b, (short)0, acc, false, false);
    }
    const int rbase = row0 + ((lane >> 4) << 3);
#pragma unroll
    for (int r = 0; r < 8; ++r)
        h2[(size_t)(rbase + r) * 16 + m] = acc[r];
}

// ---------------- attention logits ----------------
__global__ void alpha1_kernel(const float* __restrict__ h1, const float* __restrict__ a_s,
                              const float* __restrict__ a_d,
                              float* __restrict__ as1, float* __restrict__ ad1) {
    int t = blockIdx.x * blockDim.x + threadIdx.x;      // (node, head)
    if (t >= N_NODES * NHEAD1) return;
    int n = t >> 3, h = t & 7;
    float ss = 0.f, dd = 0.f;
#pragma unroll
    for (int c = 0; c < CH1; ++c) {
        float hv = h1[(size_t)n * 64 + h * 8 + c];
        ss += hv * a_s[h * 8 + c];
        dd += hv * a_d[h * 8 + c];
    }
    as1[t] = ss; ad1[t] = dd;
}
__global__ void alpha2_kernel(const float* __restrict__ h2, const float* __restrict__ a_s,
                              const float* __restrict__ a_d,
                              float* __restrict__ as2, float* __restrict__ ad2) {
    int n = blockIdx.x * blockDim.x + threadIdx.x;
    if (n >= N_NODES) return;
    float ss = 0.f, dd = 0.f;
#pragma unroll
    for (int c = 0; c < NCLS; ++c) {
        float hv = h2[(size_t)n * 16 + c];
        ss += hv * a_s[c];
        dd += hv * a_d[c];
    }
    as2[n] = ss; ad2[n] = dd;
}

// ---------------- edge helpers ----------------
__device__ __forceinline__ void edge_sd(const int* ei, int e, int& s, int& d) {
    if (e < E_EDGES) { s = ei[e]; d = ei[E_EDGES + e]; }
    else             { s = d = e - E_EDGES; }            // appended self loop
}

// layer 1: segment max (thread per edge, 8 heads)
__global__ void emax1_kernel(const int* __restrict__ ei, const float* __restrict__ as1,
                             const float* __restrict__ ad1, unsigned* __restrict__ m1) {
    int e = blockIdx.x * blockDim.x + threadIdx.x;
    if (e >= ET) return;
    int s, d; edge_sd(ei, e, s, d);
#pragma unroll
    for (int h = 0; h < NHEAD1; ++h) {
        float v = lrelu(as1[s * 8 + h] + ad1[d * 8 + h]);
        atomicMax(&m1[d * 8 + h], f2ord(v));
    }
}
// layer 1: segment sum of exp
__global__ void esum1_kernel(const int* __restrict__ ei, const float* __restrict__ as1,
                             const float* __restrict__ ad1, const unsigned* __restrict__ m1,
                             float* __restrict__ s1) {
    int e = blockIdx.x * blockDim.x + threadIdx.x;
    if (e >= ET) return;
    int s, d; edge_sd(ei, e, s, d);
#pragma unroll
    for (int h = 0; h < NHEAD1; ++h) {
        float v = lrelu(as1[s * 8 + h] + ad1[d * 8 + h]);
        atomicAdd(&s1[d * 8 + h], __expf(v - ord2f(m1[d * 8 + h])));
    }
}
// layer 1 aggregate: one wave32 per edge; lane handles channels 2l,2l+1 (head = l>>2)
__global__ void agg1_kernel(const int* __restrict__ ei, const float* __restrict__ h1,
                            const float* __restrict__ as1, const float* __restrict__ ad1,
                            const unsigned* __restrict__ m1, const float* __restrict__ s1,
                            float* __restrict__ agg1) {
    int gw = (blockIdx.x * blockDim.x + threadIdx.x) >> 5;
    int lane = threadIdx.x & 31;
    if (gw >= ET) return;
    int s, d; edge_sd(ei, gw, s, d);
    int hd = lane >> 2;
    float v = lrelu(as1[s * 8 + hd] + ad1[d * 8 + hd]);
    float w = __expf(v - ord2f(m1[d * 8 + hd])) / (s1[d * 8 + hd] + 1e-16f);
    int c = lane * 2;
    const float2 hv = *reinterpret_cast<const float2*>(&h1[(size_t)s * 64 + c]);
    atomicAdd(&agg1[(size_t)d * 64 + c],     hv.x * w);
    atomicAdd(&agg1[(size_t)d * 64 + c + 1], hv.y * w);
}
// bias + ELU in place
__global__ void elu_kernel(float* __restrict__ a, const float* __restrict__ b1) {
    int i = blockIdx.x * blockDim.x + threadIdx.x;
    if (i >= N_NODES * D1) return;
    float v = a[i] + b1[i & 63];
    a[i] = v > 0.f ? v : (__expf(v) - 1.f);
}

// layer 2 edge passes (1 head)
__global__ void emax2_kernel(const int* __restrict__ ei, const float* __restrict__ as2,
                             const float* __restrict__ ad2, unsigned* __restrict__ m2) {
    int e = blockIdx.x * blockDim.x + threadIdx.x;
    if (e >= ET) return;
    int s, d; edge_sd(ei, e, s, d);
    atomicMax(&m2[d], f2ord(lrelu(as2[s] + ad2[d])));
}
__global__ void esum2_kernel(const int* __restrict__ ei, const float* __restrict__ as2,
                             const float* __restrict__ ad2, const unsigned* __restrict__ m2,
                             float* __restrict__ s2) {
    int e = blockIdx.x * blockDim.x + threadIdx.x;
    if (e >= ET) return;
    int s, d; edge_sd(ei, e, s, d);
    atomicAdd(&s2[d], __expf(lrelu(as2[s] + ad2[d]) - ord2f(m2[d])));
}
// layer 2 aggregate: thread per (edge, class)
__global__ void agg2_kernel(const int* __restrict__ ei, const float* __restrict__ h2,
                            const float* __restrict__ as2, const float* __restrict__ ad2,
                            const unsigned* __restrict__ m2, const float* __restrict__ s2,
                            float* __restrict__ dout) {
    long long t = (long long)blockIdx.x * blockDim.x + threadIdx.x;
    if (t >= (long long)ET * NCLS) return;
    int e = (int)(t >> 4), c = (int)(t & 15);
    int s, d; edge_sd(ei, e, s, d);
    float v = lrelu(as2[s] + ad2[d]);
    float w = __expf(v - ord2f(m2[d])) / (s2[d] + 1e-16f);
    atomicAdd(&dout[(size_t)d * 16 + c], h2[(size_t)s * 16 + c] * w);
}
// bias + log_softmax in place (thread per node)
__global__ void finalize_kernel(float* __restrict__ out, const float* __restrict__ b2) {
    int n = blockIdx.x * blockDim.x + threadIdx.x;
    if (n >= N_NODES) return;
    float v[NCLS], mx = -3.4e38f;
#pragma unroll
    for (int c = 0; c < NCLS; ++c) { v[c] = out[(size_t)n * 16 + c] + b2[c]; mx = fmaxf(mx, v[c]); }
    float sum = 0.f;
#pragma unroll
    for (int c = 0; c < NCLS; ++c) sum += __expf(v[c] - mx);
    float lg = __logf(sum);
#pragma unroll
    for (int c = 0; c < NCLS; ++c) out[(size_t)n * 16 + c] = v[c] - mx - lg;
}

// ---------------- host ----------------
extern "C" void kernel_launch(void* const* d_in, const int* in_sizes, int n_in,
                              void* d_out, int out_size, void* d_ws, size_t ws_size,
                              hipStream_t stream) {
    const float* x   = (const float*)d_in[0];
    const int*   ei  = (const int*)  d_in[1];
    const float* W1  = (const float*)d_in[2];
    const float* aS1 = (const float*)d_in[3];
    const float* aD1 = (const float*)d_in[4];
    const float* b1  = (const float*)d_in[5];
    const float* W2  = (const float*)d_in[6];
    const float* aS2 = (const float*)d_in[7];
    const float* aD2 = (const float*)d_in[8];
    const float* b2  = (const float*)d_in[9];
    float* dout = (float*)d_out;

    // workspace carve-up (~36 MB of floats)
    float*    ws   = (float*)d_ws;
    float*    h1   = ws;                                     // N*64
    float*    as1  = h1  + (size_t)N_NODES * 64;             // N*8
    float*    ad1  = as1 + (size_t)N_NODES * 8;              // N*8
    unsigned* m1   = (unsigned*)(ad1 + (size_t)N_NODES * 8); // N*8
    float*    s1   = (float*)(m1 + (size_t)N_NODES * 8);     // N*8
    float*    agg1 = s1  + (size_t)N_NODES * 8;              // N*64 (becomes out1)
    float*    h2   = agg1 + (size_t)N_NODES * 64;            // N*16
    float*    as2  = h2  + (size_t)N_NODES * 16;             // N
    float*    ad2  = as2 + (size_t)N_NODES;                  // N
    unsigned* m2   = (unsigned*)(ad2 + (size_t)N_NODES);     // N
    float*    s2   = (float*)(m2 + (size_t)N_NODES);         // N

    const int B = 256;
    const int tiles_blocks = ((N_NODES / 16) + 7) / 8;       // 8 waves/block

    init1_kernel <<<(N_NODES * 64 + B - 1) / B, B, 0, stream>>>(agg1, m1, s1);
    gemm1_wmma   <<<tiles_blocks, B, 0, stream>>>(x, W1, h1);
    alpha1_kernel<<<(N_NODES * 8 + B - 1) / B, B, 0, stream>>>(h1, aS1, aD1, as1, ad1);
    emax1_kernel <<<(ET + B - 1) / B, B, 0, stream>>>(ei, as1, ad1, m1);
    esum1_kernel <<<(ET + B - 1) / B, B, 0, stream>>>(ei, as1, ad1, m1, s1);
    agg1_kernel  <<<(ET * 32 + B - 1) / B, B, 0, stream>>>(ei, h1, as1, ad1, m1, s1, agg1);
    elu_kernel   <<<(N_NODES * 64 + B - 1) / B, B, 0, stream>>>(agg1, b1);

    gemm2_wmma   <<<tiles_blocks, B, 0, stream>>>(agg1, W2, h2);
    alpha2_kernel<<<(N_NODES + B - 1) / B, B, 0, stream>>>(h2, aS2, aD2, as2, ad2);
    init2_kernel <<<(N_NODES * 16 + B - 1) / B, B, 0, stream>>>(dout, m2, s2);
    emax2_kernel <<<(ET + B - 1) / B, B, 0, stream>>>(ei, as2, ad2, m2);
    esum2_kernel <<<(ET + B - 1) / B, B, 0, stream>>>(ei, as2, ad2, m2, s2);
    agg2_kernel  <<<((long long)ET * 16 + B - 1) / B, B, 0, stream>>>(ei, h2, as2, ad2, m2, s2, dout);
    finalize_kernel<<<(N_NODES + B - 1) / B, B, 0, stream>>>(dout, b2);
}